// HomoLoss_75935021793522
// MI455X (gfx1250) — compile-verified
//
#include <hip/hip_runtime.h>
#include <hip/hip_bf16.h>

// ---------------------------------------------------------------------------
// HomoLoss on MI455X (gfx1250, wave32).
// Phase 1: bitmaps + masked-edge compaction (bandwidth-bound scan, ~19 MB).
// Phase 2: per-16-edge batch, V_WMMA_F32_16X16X4_F32 Gram chains give
//          diag(Xu Xv^T), diag(Xu Xu^T), diag(Xv Xv^T) -> cosine hinge sum.
// ---------------------------------------------------------------------------

typedef float v2f __attribute__((ext_vector_type(2)));
typedef float v8f __attribute__((ext_vector_type(8)));

#define D_FEAT 128
#define HOMO_BOOST_THRD 0.5f
#define HOMO_LOSS_WEIGHT 100.0f
#define COS_EPS 1e-8f

// ---- Phase 0: zero counter, bitmaps, output scalar -------------------------
__global__ void hl_init_kernel(unsigned* counter, unsigned* bkd_bits,
                               unsigned* tgt_bits, float* out, int nw) {
  int i = blockIdx.x * blockDim.x + threadIdx.x;
  if (i == 0) { *counter = 0u; *out = 0.0f; }
  if (i < nw) { bkd_bits[i] = 0u; tgt_bits[i] = 0u; }
}

// ---- Phase 1a: scatter membership bitmaps ----------------------------------
__global__ void hl_set_bits_kernel(const int* __restrict__ tgt, int nt,
                                   const int* __restrict__ bkd, int nb,
                                   unsigned* tgt_bits, unsigned* bkd_bits) {
  int i = blockIdx.x * blockDim.x + threadIdx.x;
  if (i < nt) {
    int n = tgt[i];
    atomicOr(&tgt_bits[n >> 5], 1u << (n & 31));
  } else if (i < nt + nb) {
    int n = bkd[i - nt];
    atomicOr(&bkd_bits[n >> 5], 1u << (n & 31));
  }
}

// ---- Phase 1b: scan all edges, compact survivors ---------------------------
__global__ void hl_scan_kernel(const int* __restrict__ eu,
                               const int* __restrict__ ev,
                               const float* __restrict__ w,
                               const unsigned* __restrict__ bkd_bits,
                               const unsigned* __restrict__ tgt_bits,
                               unsigned* counter,
                               int* __restrict__ cu, int* __restrict__ cv,
                               int nE, unsigned cap) {
  int stride = gridDim.x * blockDim.x;
  for (int e = blockIdx.x * blockDim.x + threadIdx.x; e < nE; e += stride) {
    int u = eu[e], v = ev[e];
    float ww = w[e];
    unsigned bu = (bkd_bits[u >> 5] >> (u & 31)) & 1u;
    unsigned tv = (tgt_bits[v >> 5] >> (v & 31)) & 1u;
    unsigned bv = (bkd_bits[v >> 5] >> (v & 31)) & 1u;
    unsigned tu = (tgt_bits[u >> 5] >> (u & 31)) & 1u;
    if ((ww != 0.0f) && (((bu & tv) | (bv & tu)) != 0u)) {
      unsigned p = atomicAdd(counter, 1u);
      if (p < cap) { cu[p] = u; cv[p] = v; }
    }
  }
}

// ---- Phase 2: WMMA cosine over compacted batches of 16 edges ---------------
// f32 16x16x4 WMMA operand layout (wave32): lane l (row = l&15, hi = l>>4)
// holds the float2 x_row[k0 + 2*hi .. k0 + 2*hi + 1] for BOTH the A (16x4)
// and B (4x16) roles, so one b64 load per lane per matrix per k-chunk.
__global__ void hl_cosine_kernel(const float* __restrict__ x,
                                 const int* __restrict__ cu,
                                 const int* __restrict__ cv,
                                 const unsigned* __restrict__ counter,
                                 unsigned cap, float* out) {
  int lane = threadIdx.x & 31;
  int row  = lane & 15;
  int hi   = lane >> 4;

  int count = (int)min(*counter, cap);
  if (count <= 0) return;
  int nbatches = (count + 15) >> 4;

  int wave_id = (blockIdx.x * blockDim.x + threadIdx.x) >> 5;
  int nwaves  = (gridDim.x * blockDim.x) >> 5;

  for (int b = wave_id; b < nbatches; b += nwaves) {   // wave-uniform loop
    int base = b << 4;
    int e    = base + row;
    int ec   = (e < count) ? e : (count - 1);          // clamp, mask later
    int u = cu[ec];
    int v = cv[ec];
    const float* pu = x + (size_t)u * D_FEAT + 2 * hi;
    const float* pv = x + (size_t)v * D_FEAT + 2 * hi;

    v8f acc_uv = {0.f, 0.f, 0.f, 0.f, 0.f, 0.f, 0.f, 0.f};
    v8f acc_uu = acc_uv;
    v8f acc_vv = acc_uv;

#pragma unroll 8
    for (int k0 = 0; k0 < D_FEAT; k0 += 4) {
      v2f au = *(const v2f*)(pu + k0);
      v2f av = *(const v2f*)(pv + k0);
      acc_uv = __builtin_amdgcn_wmma_f32_16x16x4_f32(
          false, au, false, av, (short)0, acc_uv, false, false);
      acc_uu = __builtin_amdgcn_wmma_f32_16x16x4_f32(
          false, au, false, au, (short)0, acc_uu, false, false);
      acc_vv = __builtin_amdgcn_wmma_f32_16x16x4_f32(
          false, av, false, av, (short)0, acc_vv, false, false);
    }

    // Diagonal of 16x16 f32 D: (i,i) for i<8 -> lane i, vgpr i;
    // for i>=8 -> lane i+16 (i.e. lanes 24..31), vgpr i-8.
    int dsel = (lane < 8) ? lane : ((lane >= 24) ? (lane - 24) : -1);
    float duv = 0.f, duu = 0.f, dvv = 0.f;
#pragma unroll
    for (int j = 0; j < 8; ++j) {
      if (dsel == j) { duv = acc_uv[j]; duu = acc_uu[j]; dvv = acc_vv[j]; }
    }

    int ei     = (lane < 8) ? lane : (lane - 16);      // edge index in batch
    bool valid = (dsel >= 0) && ((base + ei) < count);

    float nu  = fmaxf(__builtin_sqrtf(duu), COS_EPS);
    float nv  = fmaxf(__builtin_sqrtf(dvv), COS_EPS);
    float sim = duv / (nu * nv);
    float pe  = valid ? fmaxf(HOMO_BOOST_THRD - sim, 0.0f) : 0.0f;

    // wave32 reduction
    for (int off = 16; off > 0; off >>= 1)
      pe += __shfl_down(pe, off, 32);
    if (lane == 0)
      atomicAdd(out, HOMO_LOSS_WEIGHT * pe);
  }
}

// ---------------------------------------------------------------------------
extern "C" void kernel_launch(void* const* d_in, const int* in_sizes, int n_in,
                              void* d_out, int out_size, void* d_ws, size_t ws_size,
                              hipStream_t stream) {
  const int*   edge_index   = (const int*)d_in[0];   // [2, E] flat
  const float* edge_weights = (const float*)d_in[1]; // [E]
  const float* x            = (const float*)d_in[2]; // [N, 128]
  const int*   target_nodes = (const int*)d_in[3];
  const int*   bkd_nodes    = (const int*)d_in[4];

  const int nE      = in_sizes[1];
  const int nNodes  = in_sizes[2] / D_FEAT;
  const int nTarget = in_sizes[3];
  const int nBkd    = in_sizes[4];

  const int* eu = edge_index;
  const int* ev = edge_index + nE;

  // Workspace layout: [counter(16B pad)][bkd_bits][tgt_bits][cu[cap]][cv[cap]]
  const int nw    = (nNodes + 31) / 32;
  const int nwPad = ((nw + 3) / 4) * 4;                 // 16B-align sections
  char* ws = (char*)d_ws;
  unsigned* counter  = (unsigned*)ws;
  unsigned* bkd_bits = (unsigned*)(ws + 16);
  unsigned* tgt_bits = bkd_bits + nwPad;
  size_t fixed = 16 + (size_t)2 * nwPad * 4;
  size_t capS  = (ws_size > fixed) ? (ws_size - fixed) / (2 * sizeof(int)) : 0;
  unsigned cap = (unsigned)((capS > (size_t)nE) ? (size_t)nE : capS);
  int* cu = (int*)(ws + fixed);
  int* cv = cu + cap;

  float* out = (float*)d_out;

  // Phase 0: init
  {
    int threads = 256, blocks = (nw + threads - 1) / threads;
    if (blocks < 1) blocks = 1;
    hl_init_kernel<<<blocks, threads, 0, stream>>>(counter, bkd_bits, tgt_bits,
                                                   out, nw);
  }
  // Phase 1a: bitmaps
  {
    int total = nTarget + nBkd;
    int threads = 256, blocks = (total + threads - 1) / threads;
    hl_set_bits_kernel<<<blocks, threads, 0, stream>>>(
        target_nodes, nTarget, bkd_nodes, nBkd, tgt_bits, bkd_bits);
  }
  // Phase 1b: scan + compact
  {
    int threads = 256;
    int blocks = (nE + threads - 1) / threads;
    if (blocks > 8192) blocks = 8192;
    hl_scan_kernel<<<blocks, threads, 0, stream>>>(
        eu, ev, edge_weights, bkd_bits, tgt_bits, counter, cu, cv, nE, cap);
  }
  // Phase 2: WMMA cosine + reduce
  {
    int threads = 256;         // 8 waves per block
    int blocks = 64;           // 512 waves grid-striding over batches
    hl_cosine_kernel<<<blocks, threads, 0, stream>>>(x, cu, cv, counter, cap,
                                                     out);
  }
}